// SobelGradientMagnitude_59614146068864
// MI455X (gfx1250) — compile-verified
//
#include <hip/hip_runtime.h>
#include <hip/hip_bf16.h>
#include <stdint.h>
#include <math.h>

// ---- problem geometry (fixed by reference: x = (1,1,160,320,320) f32) ----
static constexpr int Dz  = 160;
static constexpr int Hy  = 320;
static constexpr int Wxc = 320;

// ---- tiling ----
static constexpr int TW      = 64;            // output tile width
static constexpr int TH      = 16;            // output tile height
static constexpr int CHUNK   = 20;            // output planes per block
static constexpr int NP      = CHUNK + 2;     // input planes streamed per block (22, even)
static constexpr int HCOLS   = TW + 2;        // 66  halo cols
static constexpr int HROWS   = TH + 2;        // 18  halo rows
static constexpr int HALO_N  = HCOLS * HROWS; // 1188 halo elements / plane
static constexpr int NTH     = 256;           // 8 wave32s
static constexpr int LD_IT   = 5;             // ceil(1188/256) -> uniform 5 async issues/wave
static constexpr int RAWBUF  = LD_IT * NTH;   // 1280 floats per LDS buffer (slack = dummy slots)

template <int N> struct IC { static constexpr int v = N; };

// CDNA5 async global->LDS copy, GV mode: mem addr = VGPR64 + ioffset.
// VDST vgpr = per-lane LDS byte address. Tracked by ASYNCcnt.
__device__ __forceinline__ void async_load_b32(unsigned lds_off,
                                               unsigned long long gaddr) {
  asm volatile("global_load_async_to_lds_b32 %0, %1, off"
               :: "v"(lds_off), "v"(gaddr) : "memory");
}

__global__ __launch_bounds__(NTH) void
SobelGradientMagnitude_59614146068864_kernel(const float* __restrict__ in,
                                             float* __restrict__ out) {
  __shared__ float raw[2][RAWBUF];

  const int tid = threadIdx.x;
  const int w   = tid & (TW - 1);        // 0..63
  const int h0  = (tid >> 6) << 2;       // 0,4,8,12 : thread owns rows h0..h0+3
  const int w0t = blockIdx.x * TW;
  const int h0t = blockIdx.y * TH;
  const int z0  = blockIdx.z * CHUNK;

  // ---- per-element halo-load descriptors (z-invariant, computed once) ----
  unsigned ldsOff[LD_IT];   // LDS byte address (buffer 0)
  unsigned hwByte[LD_IT];   // clamped (h,w) byte offset inside a plane
  unsigned elIdx [LD_IT];   // float index in raw buffer (for zero fixup)
  bool     oobFix[LD_IT];   // real element whose (h,w) is outside the volume
  bool     realEl[LD_IT];

  const unsigned ldsBase = (unsigned)(uintptr_t)(&raw[0][0]);
#pragma unroll
  for (int i = 0; i < LD_IT; ++i) {
    const int e = i * NTH + tid;                  // 0..1279
    elIdx[i]  = (unsigned)e;
    ldsOff[i] = ldsBase + (unsigned)e * 4u;
    const bool real = e < HALO_N;
    realEl[i] = real;
    const int hr = e / HCOLS;
    const int wc = e - hr * HCOLS;
    const int gh = h0t - 1 + hr;
    const int gww = w0t - 1 + wc;
    const bool oob = (gh < 0) | (gh >= Hy) | (gww < 0) | (gww >= Wxc);
    const int hcl = gh  < 0 ? 0 : (gh  >= Hy  ? Hy  - 1 : gh);
    const int wcl = gww < 0 ? 0 : (gww >= Wxc ? Wxc - 1 : gww);
    hwByte[i] = real ? (unsigned)((hcl * Wxc + wcl) * 4) : 0u;  // dummies load plane[0,0]
    oobFix[i] = real & oob;
  }

  // rolling accumulators for the two in-flight output planes (parity-indexed)
  float Sgx[2][4], Sgy[2][4], Sgz[2][4];
#pragma unroll
  for (int q = 0; q < 2; ++q)
#pragma unroll
    for (int r = 0; r < 4; ++r) { Sgx[q][r] = 0.f; Sgy[q][r] = 0.f; Sgz[q][r] = 0.f; }

  const unsigned long long inAddr = (unsigned long long)(uintptr_t)in;
  const unsigned long long planeBytes = (unsigned long long)Hy * Wxc * 4ull;

  // ---- prologue: async-load plane p=0 (z = z0-1, clamped) into buffer 0 ----
  {
    const int zn = z0 - 1;
    const int zcl = zn < 0 ? 0 : zn;
    const unsigned long long sb = inAddr + (unsigned long long)zcl * planeBytes;
#pragma unroll
    for (int i = 0; i < LD_IT; ++i)
      async_load_b32(ldsOff[i], sb + (unsigned long long)hwByte[i]);
  }

  auto step = [&](auto qc, int p) {
    constexpr int Q = decltype(qc)::v;         // p & 1, static
    const int zp = z0 - 1 + p;                 // input plane handled this step

    // 1) issue async loads for plane p+1 into buffer !Q, then wait for plane p
    if (p + 1 < NP) {
      const int zn  = zp + 1;
      const int zcl = zn < 0 ? 0 : (zn >= Dz ? Dz - 1 : zn);
      const unsigned long long sb = inAddr + (unsigned long long)zcl * planeBytes;
      constexpr unsigned bufadd = (Q ^ 1) ? (unsigned)(RAWBUF * 4) : 0u;
#pragma unroll
      for (int i = 0; i < LD_IT; ++i)
        async_load_b32(ldsOff[i] + bufadd, sb + (unsigned long long)hwByte[i]);
      asm volatile("s_wait_asynccnt 0x5" ::: "memory");   // plane p resident, p+1 in flight
    } else {
      asm volatile("s_wait_asynccnt 0x0" ::: "memory");
    }

    // 2) zero-fixup: emulate 'SAME' zero padding at volume edges
    {
      const bool zoob = (zp < 0) || (zp >= Dz);
      float* bufp = &raw[Q][0];
#pragma unroll
      for (int i = 0; i < LD_IT; ++i)
        if (oobFix[i] || (zoob && realEl[i])) bufp[elIdx[i]] = 0.0f;
    }
    __syncthreads();

    // 3) separable in-plane reduction + cross-plane accumulation
    {
      const float* bufp = &raw[Q][0];
      float sw[6], dw[6];                       // rows h0-1 .. h0+4 (halo coords)
#pragma unroll
      for (int rr = 0; rr < 6; ++rr) {
        const int base = (h0 + rr) * HCOLS + w; // col w in halo space == global w-1
        const float m  = bufp[base];
        const float c0 = bufp[base + 1];
        const float pl = bufp[base + 2];
        sw[rr] = fmaf(2.0f, c0, m + pl);        // [1,2,1] along W
        dw[rr] = m - pl;                        // [1,0,-1] along W
      }
#pragma unroll
      for (int r = 0; r < 4; ++r) {
        const float av = fmaf(2.0f, sw[r + 1], sw[r] + sw[r + 2]); // Sh·Sw
        const float bv = sw[r] - sw[r + 2];                        // Dh·Sw
        const float cv = fmaf(2.0f, dw[r + 1], dw[r] + dw[r + 2]); // Sh·Dw

        if (p >= 2) {  // finalize output plane zp-1 (accumulator opened at p-2)
          const float gx = Sgx[Q][r] - av;                 // a(z-1) - a(z+1)
          const float gy = Sgy[Q][r] + bv;                 // Sd of b
          const float gz = Sgz[Q][r] + cv;                 // Sd of c
          const float mag =
              sqrtf(fmaf(gx, gx, fmaf(gy, gy, fmaf(gz, gz, 1e-7f))));
          const size_t oi = ((size_t)(zp - 1) * Hy + (size_t)(h0t + h0 + r)) * Wxc
                            + (size_t)(w0t + w);
          out[oi] = mag;
        }
        // center contribution to output plane zp (other parity): weight 2, gx weight 0
        Sgy[Q ^ 1][r] += 2.0f * bv;
        Sgz[Q ^ 1][r] += 2.0f * cv;
        // open accumulator for output plane zp+1 (this parity): weight +1
        Sgx[Q][r] = av;
        Sgy[Q][r] = bv;
        Sgz[Q][r] = cv;
      }
    }
    __syncthreads();   // all reads of raw[Q] done before it is refilled at step p+1
  };

#pragma unroll 1
  for (int pp = 0; pp < NP; pp += 2) {   // NP == 22, even
    step(IC<0>{}, pp);
    step(IC<1>{}, pp + 1);
  }
}

extern "C" void kernel_launch(void* const* d_in, const int* in_sizes, int n_in,
                              void* d_out, int out_size, void* d_ws, size_t ws_size,
                              hipStream_t stream) {
  (void)in_sizes; (void)n_in; (void)d_ws; (void)ws_size; (void)out_size;
  const float* x = (const float*)d_in[0];
  float* out = (float*)d_out;
  dim3 grid(Wxc / TW, Hy / TH, Dz / CHUNK);   // (5, 20, 8) = 800 blocks
  dim3 block(NTH);
  SobelGradientMagnitude_59614146068864_kernel<<<grid, block, 0, stream>>>(x, out);
}